// GCN_9259949490539
// MI455X (gfx1250) — compile-verified
//
#include <hip/hip_runtime.h>

// ---------------------------------------------------------------------------
// 2-layer GCN for MI455X (gfx1250, wave32).
//  - Dense linears use V_WMMA_F32_16X16X4_F32 (exact f32 semantics vs ref).
//  - Edge aggregation is L2-resident scatter-add (features: 12.8 MB << 192 MB L2).
//  - dinv scaling folded into WMMA epilogues; self-loop term seeds the
//    accumulators so no extra zero/self-loop passes are needed.
// ---------------------------------------------------------------------------

typedef float v2f __attribute__((ext_vector_type(2)));
typedef float v8f __attribute__((ext_vector_type(8)));

// ---- degree / normalization -----------------------------------------------

__global__ void k_init_deg(float* __restrict__ deg, int N) {
  int i = blockIdx.x * blockDim.x + threadIdx.x;
  if (i < N) deg[i] = 1.0f;  // self loop
}

__global__ void k_deg_accum(const int* __restrict__ dst, float* __restrict__ deg, int E) {
  int e = blockIdx.x * blockDim.x + threadIdx.x;
  if (e < E) atomicAdd(&deg[dst[e]], 1.0f);
}

__global__ void k_dinv(float* __restrict__ deg, int N) {
  int i = blockIdx.x * blockDim.x + threadIdx.x;
  if (i < N) deg[i] = rsqrtf(deg[i]);  // deg >= 1 always (self loop)
}

// ---- layer 1: h1s = (x @ W1) * dinv[row];  agg1 seeded with self-loop -----
// One wave per 16-row tile. A: 16x16 tile of x, K consumed 4 per WMMA.
// Two N-tiles (32 output features).

__global__ void k_linear1(const float* __restrict__ x, const float* __restrict__ W1,
                          const float* __restrict__ dinv, float* __restrict__ h1s,
                          float* __restrict__ agg1, int nTiles) {
  int wave = threadIdx.x >> 5;
  int tile = blockIdx.x * (blockDim.x >> 5) + wave;
  if (tile >= nTiles) return;  // wave-uniform: EXEC stays all-ones for WMMA

  int lane = threadIdx.x & 31;
  int n    = lane & 15;   // N index for B/C/D
  int half = lane >> 4;   // lane group
  int m    = lane & 15;   // M index for A
  int row0 = tile * 16;

  const float* xrow = x + (size_t)(row0 + m) * 16;
  v8f c0 = {}; v8f c1 = {};
#pragma unroll
  for (int k = 0; k < 4; ++k) {
    int kk = 4 * k + 2 * half;                 // VGPR v holds K = kk + v
    v2f a  = { xrow[kk], xrow[kk + 1] };
    v2f b0 = { W1[kk * 32 + n],       W1[(kk + 1) * 32 + n] };
    v2f b1 = { W1[kk * 32 + 16 + n],  W1[(kk + 1) * 32 + 16 + n] };
    c0 = __builtin_amdgcn_wmma_f32_16x16x4_f32(false, a, false, b0, (short)0, c0, false, false);
    c1 = __builtin_amdgcn_wmma_f32_16x16x4_f32(false, a, false, b1, (short)0, c1, false, false);
  }
#pragma unroll
  for (int r = 0; r < 8; ++r) {
    int row = row0 + r + 8 * half;             // C/D: M = r + 8*half, N = n
    float s  = dinv[row];
    float v0 = c0[r] * s;
    float v1 = c1[r] * s;
    size_t o = (size_t)row * 32;
    h1s[o + n]       = v0;  agg1[o + n]       = v0;   // agg seeded w/ self loop
    h1s[o + 16 + n]  = v1;  agg1[o + 16 + n]  = v1;
  }
}

// ---- edge scatter layer 1: agg1[dst] += h1s[src], 32 feats/edge -----------
// 32 consecutive lanes = one edge -> coalesced gather + coalesced atomics.

__global__ void k_scatter1(const int* __restrict__ src, const int* __restrict__ dst,
                           const float* __restrict__ h1s, float* __restrict__ agg1, int E) {
  int tid = blockIdx.x * blockDim.x + threadIdx.x;
  int e = tid >> 5;
  int f = tid & 31;
  if (e < E) {
    int s = src[e], d = dst[e];
    atomicAdd(&agg1[(size_t)d * 32 + f], h1s[(size_t)s * 32 + f]);
  }
}

// ---- layer 2: z = relu(dinv*agg1 + b1); h2s = (z @ W2pad) * dinv ----------
// K = 32 (8 WMMA steps). W2 is 32x3, zero-padded to N=16; only n<3 stored.

__global__ void k_linear2(const float* __restrict__ agg1, const float* __restrict__ W2,
                          const float* __restrict__ b1, const float* __restrict__ dinv,
                          float* __restrict__ h2s, float* __restrict__ agg2, int nTiles) {
  int wave = threadIdx.x >> 5;
  int tile = blockIdx.x * (blockDim.x >> 5) + wave;
  if (tile >= nTiles) return;

  int lane = threadIdx.x & 31;
  int n    = lane & 15;
  int half = lane >> 4;
  int m    = lane & 15;
  int row0 = tile * 16;

  float sA = dinv[row0 + m];
  const float* arow = agg1 + (size_t)(row0 + m) * 32;
  v8f c = {};
#pragma unroll
  for (int k = 0; k < 8; ++k) {
    int kk = 4 * k + 2 * half;
    float z0 = fmaxf(fmaf(sA, arow[kk],     b1[kk]),     0.0f);
    float z1 = fmaxf(fmaf(sA, arow[kk + 1], b1[kk + 1]), 0.0f);
    v2f a = { z0, z1 };
    v2f b = { (n < 3) ? W2[kk * 3 + n] : 0.0f,
              (n < 3) ? W2[(kk + 1) * 3 + n] : 0.0f };
    c = __builtin_amdgcn_wmma_f32_16x16x4_f32(false, a, false, b, (short)0, c, false, false);
  }
  if (n < 3) {
#pragma unroll
    for (int r = 0; r < 8; ++r) {
      int row = row0 + r + 8 * half;
      float v = c[r] * dinv[row];
      h2s[(size_t)row * 3 + n]  = v;
      agg2[(size_t)row * 3 + n] = v;                 // self-loop seed
    }
  }
}

// ---- edge scatter layer 2: 3 feats/edge (4 lanes/edge, lane 3 idle) -------

__global__ void k_scatter2(const int* __restrict__ src, const int* __restrict__ dst,
                           const float* __restrict__ h2s, float* __restrict__ agg2, int E) {
  int tid = blockIdx.x * blockDim.x + threadIdx.x;
  int e = tid >> 2;
  int f = tid & 3;
  if (e < E && f < 3) {
    int s = src[e], d = dst[e];
    atomicAdd(&agg2[(size_t)d * 3 + f], h2s[(size_t)s * 3 + f]);
  }
}

// ---- bias + log_softmax ---------------------------------------------------

__global__ void k_final(const float* __restrict__ agg2, const float* __restrict__ dinv,
                        const float* __restrict__ b2, float* __restrict__ out, int N) {
  int i = blockIdx.x * blockDim.x + threadIdx.x;
  if (i >= N) return;
  float di = dinv[i];
  float v0 = fmaf(di, agg2[(size_t)i * 3 + 0], b2[0]);
  float v1 = fmaf(di, agg2[(size_t)i * 3 + 1], b2[1]);
  float v2 = fmaf(di, agg2[(size_t)i * 3 + 2], b2[2]);
  float mx = fmaxf(fmaxf(v0, v1), v2);
  float lse = logf(expf(v0 - mx) + expf(v1 - mx) + expf(v2 - mx));
  out[(size_t)i * 3 + 0] = v0 - mx - lse;
  out[(size_t)i * 3 + 1] = v1 - mx - lse;
  out[(size_t)i * 3 + 2] = v2 - mx - lse;
}

// ---------------------------------------------------------------------------

extern "C" void kernel_launch(void* const* d_in, const int* in_sizes, int n_in,
                              void* d_out, int out_size, void* d_ws, size_t ws_size,
                              hipStream_t stream) {
  const float* x  = (const float*)d_in[0];
  const int*   ei = (const int*)d_in[1];
  const float* W1 = (const float*)d_in[2];
  const float* b1 = (const float*)d_in[3];
  const float* W2 = (const float*)d_in[4];
  const float* b2 = (const float*)d_in[5];

  const int N = in_sizes[0] / 16;   // 100000
  const int E = in_sizes[1] / 2;    // 3200000
  const int* src = ei;
  const int* dst = ei + E;

  // workspace layout (floats): dinv[N] | h1s[N*32] | agg1[N*32] | h2s[N*3] | agg2[N*3]
  float* ws   = (float*)d_ws;
  float* dinv = ws;
  float* h1s  = dinv + (size_t)N;
  float* agg1 = h1s  + (size_t)N * 32;
  float* h2s  = agg1 + (size_t)N * 32;
  float* agg2 = h2s  + (size_t)N * 3;
  float* out  = (float*)d_out;

  const int T = 256;
  int nTiles   = (N + 15) / 16;                 // 6250 (exact)
  int gTiles   = (nTiles + 7) / 8;              // 8 waves per 256-thread block
  int gN       = (N + T - 1) / T;
  int gE       = (E + T - 1) / T;
  long long s1 = (long long)E * 32;
  long long s2 = (long long)E * 4;
  int gS1      = (int)((s1 + T - 1) / T);
  int gS2      = (int)((s2 + T - 1) / T);

  k_init_deg <<<gN, T, 0, stream>>>(dinv, N);
  k_deg_accum<<<gE, T, 0, stream>>>(dst, dinv, E);
  k_dinv     <<<gN, T, 0, stream>>>(dinv, N);

  k_linear1  <<<gTiles, T, 0, stream>>>(x, W1, dinv, h1s, agg1, nTiles);
  k_scatter1 <<<gS1, T, 0, stream>>>(src, dst, h1s, agg1, E);

  k_linear2  <<<gTiles, T, 0, stream>>>(agg1, W2, b1, dinv, h2s, agg2, nTiles);
  k_scatter2 <<<gS2, T, 0, stream>>>(src, dst, h2s, agg2, E);

  k_final    <<<gN, T, 0, stream>>>(agg2, dinv, b2, out, N);
}